// GCN_35751307772796
// MI455X (gfx1250) — compile-verified
//
#include <hip/hip_runtime.h>
#include <hip/hip_bf16.h>

// ---------------- problem constants (match reference) ----------------
static constexpr int NN = 100000;   // nodes (3125 * 32)
static constexpr int FF = 512;      // input features
static constexpr int HH = 256;      // hidden
static constexpr int CC = 64;       // classes
static constexpr int EE = 1600000;  // edges
static constexpr int KK = 10;       // hops
static constexpr float BN_EPS = 1e-5f;

typedef __attribute__((ext_vector_type(2))) float v2f;
typedef __attribute__((ext_vector_type(8))) float v8f;
typedef int v4i_ __attribute__((vector_size(16)));
typedef __attribute__((address_space(1))) v4i_* gptr4;   // global v4i*
typedef __attribute__((address_space(3))) v4i_* lptr4;   // LDS v4i*

// Optional CDNA5 async global->LDS path (ASYNCcnt-tracked).
#if defined(__has_builtin)
#if __has_builtin(__builtin_amdgcn_global_load_async_to_lds_b128) && \
    __has_builtin(__builtin_amdgcn_s_wait_asynccnt)
#define USE_ASYNC_LDS 1
#endif
#endif
#ifndef USE_ASYNC_LDS
#define USE_ASYNC_LDS 0
#endif

// =====================================================================
// GEMM: D[M,Nc] = A[M,K] @ Bt^T + bias  (fp32 WMMA 16x16x4)
//   Bt is B pre-transposed: Bt[n*K + k] = B[k*Nc + n]  (L2-resident)
//
// Block = 32-row M-super-tile; one wave per 64-column strip
//   (blockDim.x = 32 * Nc/64; Nc in {64, 256}).
// A tile (32 x KCH) staged in LDS (coalesced b128 / async-to-LDS).
// Each wave: 2 M-tiles x 4 N-tiles = 8 accumulators.
//   A fragment reused x4, B fragment (one global b64) reused x2.
//
// A frag: lane -> row M = m0 + lane%16, K pair = k + 2*(lane/16)
// B frag: lane -> col N = n0 + lane%16, K pair = k + 2*(lane/16)
// D:      vgpr r -> row M = m0 + r + 8*(lane/16), col = n0 + lane%16
// =====================================================================
static constexpr int KCH  = 64;        // K chunk staged in LDS
static constexpr int SROW = KCH + 4;   // padded row stride (dwords, 16B-mult)

__global__ void wmma_gemm_bias_f32(const float* __restrict__ A,
                                   const float* __restrict__ Bt,
                                   const float* __restrict__ bias,
                                   float* __restrict__ D,
                                   int M, int K, int Nc) {
  __shared__ float As[32 * SROW];      // 8.5 KB
  const int lane = threadIdx.x & 31;
  const int wave = threadIdx.x >> 5;
  const int half = lane >> 4;          // 0..1
  const int l16  = lane & 15;          // 0..15
  const int m0   = blockIdx.x << 5;    // 32-row super-tile
  const int n0w  = wave << 6;          // this wave's 64-column strip
  const int nthreads = blockDim.x;

  v8f acc[8] = {v8f{}, v8f{}, v8f{}, v8f{}, v8f{}, v8f{}, v8f{}, v8f{}};

  for (int kb = 0; kb < K; kb += KCH) {
    // ---- cooperative A-tile load: 32 rows x KCH floats, b128 granules ----
    for (int idx = threadIdx.x; idx < 32 * (KCH / 4); idx += nthreads) {
      const int r  = idx >> 4;         // row 0..31
      const int c4 = idx & 15;         // float4 index within row
      const float* gsrc = A + (size_t)(m0 + r) * K + kb + (c4 << 2);
      float* ldst = &As[r * SROW + (c4 << 2)];
#if USE_ASYNC_LDS
      __builtin_amdgcn_global_load_async_to_lds_b128(
          (gptr4)(uintptr_t)gsrc,
          (lptr4)(unsigned)(uintptr_t)ldst, 0, 0);
#else
      *(float4*)ldst = *(const float4*)gsrc;
#endif
    }
    // prefetch next K chunk into cache while this one computes
    if (kb + KCH < K) {
      for (int idx = threadIdx.x; idx < 32 * (KCH / 4); idx += nthreads) {
        const int r  = idx >> 4;
        const int c4 = idx & 15;
        __builtin_prefetch(A + (size_t)(m0 + r) * K + kb + KCH + (c4 << 2), 0, 1);
      }
    }
#if USE_ASYNC_LDS
    __builtin_amdgcn_s_wait_asynccnt(0);
#endif
    __syncthreads();

#pragma unroll 4
    for (int kk = 0; kk < KCH; kk += 4) {
      const int ka = kk + (half << 1);
      // A fragments for both M-tiles (bank = (4*row + ka) % 64: conflict-free)
      v2f a0 = *(const v2f*)&As[l16 * SROW + ka];
      v2f a1 = *(const v2f*)&As[(16 + l16) * SROW + ka];
#pragma unroll
      for (int g = 0; g < 4; ++g) {
        const int n = n0w + (g << 4) + l16;
        // one vector load per B fragment from transposed weights
        v2f b = *(const v2f*)(Bt + (size_t)n * K + kb + ka);
        acc[g] = __builtin_amdgcn_wmma_f32_16x16x4_f32(
            false, a0, false, b, (short)0, acc[g], false, false);
        acc[4 + g] = __builtin_amdgcn_wmma_f32_16x16x4_f32(
            false, a1, false, b, (short)0, acc[4 + g], false, false);
      }
    }
    __syncthreads();
  }

  // ---- store 8 16x16 tiles + bias ----
#pragma unroll
  for (int mt = 0; mt < 2; ++mt) {
#pragma unroll
    for (int g = 0; g < 4; ++g) {
      const int n = n0w + (g << 4) + l16;
      const float bv = bias[n];
#pragma unroll
      for (int r = 0; r < 8; ++r) {
        const int mr = m0 + (mt << 4) + (half << 3) + r;
        D[(size_t)mr * Nc + n] = acc[4 * mt + g][r] + bv;
      }
    }
  }
}

// Bt[n*K + k] = B[k*Nc + n]; writes coalesced along k.
__global__ void transpose_B(const float* __restrict__ B,
                            float* __restrict__ Bt, int K, int Nc) {
  int idx = blockIdx.x * blockDim.x + threadIdx.x;
  if (idx >= K * Nc) return;
  int n = idx / K;
  int k = idx - n * K;
  Bt[idx] = B[(size_t)k * Nc + n];
}

// ---------------- utility ----------------
__global__ void zero_f32(float* __restrict__ p, int n) {
  int i = blockIdx.x * blockDim.x + threadIdx.x;
  if (i < n) p[i] = 0.f;
}

// ---------------- BatchNorm (training-mode, biased var) ----------------
__global__ void bn_partial(const float* __restrict__ Hm, int M, int Ncols,
                           int rowsPerBlock,
                           float* __restrict__ sums, float* __restrict__ sqs) {
  const int j = threadIdx.x;                 // column (blockDim.x == Ncols)
  const int r0 = blockIdx.x * rowsPerBlock;
  const int r1 = min(r0 + rowsPerBlock, M);
  float s = 0.f, s2 = 0.f;
  for (int i = r0; i < r1; ++i) {
    float v = Hm[(size_t)i * Ncols + j];
    s += v; s2 += v * v;
  }
  atomicAdd(&sums[j], s);
  atomicAdd(&sqs[j], s2);
}

__global__ void bn_finalize(const float* __restrict__ sums,
                            const float* __restrict__ sqs, int M,
                            float* __restrict__ mu, float* __restrict__ rs) {
  int j = blockIdx.x * blockDim.x + threadIdx.x;
  float m = sums[j] / (float)M;
  float var = sqs[j] / (float)M - m * m;
  mu[j] = m;
  rs[j] = rsqrtf(var + BN_EPS);
}

__global__ void bn_apply_relu(float* __restrict__ Hm, int total, int NcolsMask,
                              const float* __restrict__ mu,
                              const float* __restrict__ rs,
                              const float* __restrict__ g,
                              const float* __restrict__ be) {
  int idx = blockIdx.x * blockDim.x + threadIdx.x;
  if (idx >= total) return;
  int j = idx & NcolsMask;
  float v = (Hm[idx] - mu[j]) * rs[j] * g[j] + be[j];
  Hm[idx] = fmaxf(v, 0.f);
}

// ---------------- graph degree / normalization ----------------
__global__ void deg_init(float* __restrict__ deg, int n) {
  int i = blockIdx.x * blockDim.x + threadIdx.x;
  if (i < n) deg[i] = 1.f;                  // self-loop contributes 1
}

__global__ void deg_edges(const int* __restrict__ col,
                          float* __restrict__ deg, int E) {
  int e = blockIdx.x * blockDim.x + threadIdx.x;
  if (e < E) atomicAdd(&deg[col[e]], 1.f);
}

__global__ void dis_kernel(const float* __restrict__ deg,
                           float* __restrict__ dis, int n) {
  int i = blockIdx.x * blockDim.x + threadIdx.x;
  if (i < n) dis[i] = rsqrtf(deg[i]);       // deg >= 1 always
}

// ---------------- propagation ----------------
__global__ void self_loop_init(const float* __restrict__ xk,
                               const float* __restrict__ dis,
                               float* __restrict__ xn, int total) {
  int idx = blockIdx.x * blockDim.x + threadIdx.x;
  if (idx >= total) return;
  float d = dis[idx >> 6];                  // C = 64
  xn[idx] = d * d * xk[idx];
}

// wave per edge: 32 lanes x float2 = 64 channels; scatter-add to target
__global__ void edge_scatter(const int* __restrict__ row,
                             const int* __restrict__ col,
                             const float* __restrict__ dis,
                             const float* __restrict__ xk,
                             float* __restrict__ xn, int E) {
  int e = blockIdx.x * (blockDim.x >> 5) + (threadIdx.x >> 5);
  if (e >= E) return;                       // wave-uniform
  const int lane = threadIdx.x & 31;
  const int r = row[e];
  const int c = col[e];
  const float nrm = dis[r] * dis[c];
  const float2 v = *(const float2*)(xk + (size_t)r * CC + lane * 2);
  float* dst = xn + (size_t)c * CC + lane * 2;
  atomicAdd(dst + 0, nrm * v.x);
  atomicAdd(dst + 1, nrm * v.y);
}

// fused adaptive score: s = sigmoid(pps_k . proj_w + proj_b);
// acc += s * pps_k  (first==1 overwrites). Wave per node.
__global__ void score_accumulate(const float* __restrict__ pps,
                                 const float* __restrict__ pw,
                                 const float* __restrict__ pb,
                                 float* __restrict__ acc, int N, int first) {
  int node = blockIdx.x * (blockDim.x >> 5) + (threadIdx.x >> 5);
  if (node >= N) return;                    // wave-uniform
  const int lane = threadIdx.x & 31;
  const float2 v = *(const float2*)(pps + (size_t)node * CC + lane * 2);
  const float2 w = *(const float2*)(pw + lane * 2);
  float p = v.x * w.x + v.y * w.y;
#pragma unroll
  for (int off = 16; off > 0; off >>= 1) p += __shfl_xor(p, off, 32);
  const float s = 1.f / (1.f + __expf(-(p + pb[0])));
  float2 a;
  if (first) {
    a.x = s * v.x; a.y = s * v.y;
  } else {
    a = *(const float2*)(acc + (size_t)node * CC + lane * 2);
    a.x += s * v.x; a.y += s * v.y;
  }
  *(float2*)(acc + (size_t)node * CC + lane * 2) = a;
}

// ---------------- log_softmax over C=64, wave per node ----------------
__global__ void log_softmax64(const float* __restrict__ acc,
                              float* __restrict__ out, int N) {
  int node = blockIdx.x * (blockDim.x >> 5) + (threadIdx.x >> 5);
  if (node >= N) return;                    // wave-uniform
  const int lane = threadIdx.x & 31;
  const float2 v = *(const float2*)(acc + (size_t)node * CC + lane * 2);
  float mx = fmaxf(v.x, v.y);
#pragma unroll
  for (int off = 16; off > 0; off >>= 1) mx = fmaxf(mx, __shfl_xor(mx, off, 32));
  float e = __expf(v.x - mx) + __expf(v.y - mx);
#pragma unroll
  for (int off = 16; off > 0; off >>= 1) e += __shfl_xor(e, off, 32);
  const float l = mx + logf(e);
  float2 o; o.x = v.x - l; o.y = v.y - l;
  *(float2*)(out + (size_t)node * CC + lane * 2) = o;
}

// =====================================================================
extern "C" void kernel_launch(void* const* d_in, const int* in_sizes, int n_in,
                              void* d_out, int out_size, void* d_ws, size_t ws_size,
                              hipStream_t stream) {
  const float* x   = (const float*)d_in[0];
  const float* W0  = (const float*)d_in[1];
  const float* b0  = (const float*)d_in[2];
  const float* g0  = (const float*)d_in[3];
  const float* be0 = (const float*)d_in[4];
  const float* W1  = (const float*)d_in[5];
  const float* b1  = (const float*)d_in[6];
  const float* g1  = (const float*)d_in[7];
  const float* be1 = (const float*)d_in[8];
  const float* W2  = (const float*)d_in[9];
  const float* b2  = (const float*)d_in[10];
  const float* pw  = (const float*)d_in[11];   // [C,1]
  const float* pb  = (const float*)d_in[12];   // scalar
  const int*   ei  = (const int*)d_in[13];     // [2,E]
  const int* row = ei;        // message sources
  const int* col = ei + EE;   // aggregation targets

  // -------- workspace layout (floats) --------
  float* ws   = (float*)d_ws;
  float* hA   = ws;                         // [N,H]
  float* hB   = hA + (size_t)NN * HH;       // [N,H]
  float* h2   = hB + (size_t)NN * HH;       // [N,C] logits / pps_0
  float* xA   = h2 + (size_t)NN * CC;       // [N,C] hop ping
  float* xB   = xA + (size_t)NN * CC;       // [N,C] hop pong
  float* accO = xB + (size_t)NN * CC;       // [N,C] weighted sum
  float* deg  = accO + (size_t)NN * CC;     // [N]
  float* dis  = deg + NN;                   // [N]
  float* mu   = dis + NN;                   // [H]
  float* rs   = mu + HH;                    // [H]
  float* sums = rs + HH;                    // [H]
  float* sqs  = sums + HH;                  // [H]
  float* Bt   = sqs + HH;                   // [max F*H] transposed weights

  const int MT32 = NN / 32;                 // 3125 super-tiles
  const int rowsPerBlock = (NN + 255) / 256;

  // ---- layer 0: x @ W0 + b0 ; BN ; ReLU ----
  {
    transpose_B<<<(FF * HH + 255) / 256, 256, 0, stream>>>(W0, Bt, FF, HH);
    wmma_gemm_bias_f32<<<MT32, 32 * (HH / 64), 0, stream>>>(x, Bt, b0, hA, NN, FF, HH);
    zero_f32<<<(2 * HH + 255) / 256, 256, 0, stream>>>(sums, 2 * HH);
    bn_partial<<<256, HH, 0, stream>>>(hA, NN, HH, rowsPerBlock, sums, sqs);
    bn_finalize<<<1, HH, 0, stream>>>(sums, sqs, NN, mu, rs);
    int tot = NN * HH;
    bn_apply_relu<<<(tot + 255) / 256, 256, 0, stream>>>(hA, tot, HH - 1, mu, rs, g0, be0);
  }

  // ---- layer 1: hA @ W1 + b1 ; BN ; ReLU ----
  {
    transpose_B<<<(HH * HH + 255) / 256, 256, 0, stream>>>(W1, Bt, HH, HH);
    wmma_gemm_bias_f32<<<MT32, 32 * (HH / 64), 0, stream>>>(hA, Bt, b1, hB, NN, HH, HH);
    zero_f32<<<(2 * HH + 255) / 256, 256, 0, stream>>>(sums, 2 * HH);
    bn_partial<<<256, HH, 0, stream>>>(hB, NN, HH, rowsPerBlock, sums, sqs);
    bn_finalize<<<1, HH, 0, stream>>>(sums, sqs, NN, mu, rs);
    int tot = NN * HH;
    bn_apply_relu<<<(tot + 255) / 256, 256, 0, stream>>>(hB, tot, HH - 1, mu, rs, g1, be1);
  }

  // ---- layer 2: hB @ W2 + b2 -> h2 [N,C] ----
  transpose_B<<<(HH * CC + 255) / 256, 256, 0, stream>>>(W2, Bt, HH, CC);
  wmma_gemm_bias_f32<<<MT32, 32 * (CC / 64), 0, stream>>>(hB, Bt, b2, h2, NN, HH, CC);

  // ---- GCN normalization: deg (with self-loops), dis = deg^-1/2 ----
  deg_init<<<(NN + 255) / 256, 256, 0, stream>>>(deg, NN);
  deg_edges<<<(EE + 255) / 256, 256, 0, stream>>>(col, deg, EE);
  dis_kernel<<<(NN + 255) / 256, 256, 0, stream>>>(deg, dis, NN);

  // ---- hop 0 score contribution (pps_0 = h2) ----
  const int nodeWaves = NN / 8;             // 8 waves/block, exact
  score_accumulate<<<nodeWaves, 256, 0, stream>>>(h2, pw, pb, accO, NN, 1);

  // ---- K hops: xk+1 = D^-1/2 (A+I) D^-1/2 xk, fused score-accumulate ----
  const float* cur = h2;
  const int ncTot = NN * CC;
  for (int k = 0; k < KK; ++k) {
    float* nxt = (k & 1) ? xB : xA;
    self_loop_init<<<(ncTot + 255) / 256, 256, 0, stream>>>(cur, dis, nxt, ncTot);
    edge_scatter<<<EE / 8, 256, 0, stream>>>(row, col, dis, cur, nxt, EE);
    score_accumulate<<<nodeWaves, 256, 0, stream>>>(nxt, pw, pb, accO, NN, 0);
    cur = nxt;
  }

  // ---- final log_softmax over classes ----
  log_softmax64<<<nodeWaves, 256, 0, stream>>>(accO, (float*)d_out, NN);
}